// LoMoEOutputHeadE2E_15977278341949
// MI455X (gfx1250) — compile-verified
//
#include <hip/hip_runtime.h>
#include <math.h>

// ---------------------------------------------------------------------------
// LoMoE output head, fp32, CDNA5 (gfx1250) wave32 + V_WMMA_F32_16X16X4_F32
// GEMM: 32x64 wave tile, split-K x4 into partial buffers (deterministic).
// ---------------------------------------------------------------------------

typedef __attribute__((ext_vector_type(2))) float v2f;   // f32 WMMA A/B frag
typedef __attribute__((ext_vector_type(8))) float v8f;   // f32 WMMA C/D frag

#define Bn 32
#define Vn 21
#define Dn 512
#define Pn 64
#define INn (Dn * Pn)          // 32768
#define OUTn 720
#define En 8
#define Rn 16
#define Hn 256
#define Mn (Bn * Vn)           // 672
#define NLORA (En * Rn)        // 128
#define NTOT (OUTn + NLORA)    // 848
#define SCALING 1.0f           // 16 / R

#define KCHUNKS 4
#define KC (INn / KCHUNKS)     // 8192 per split-K chunk
#define MT2 (Mn / 32)          // 21 (exact, no M tail)
#define NT2 ((NTOT + 63) / 64) // 14 (tail tile: only frag j=0 valid)
#define JOBS (MT2 * NT2)       // 294 per chunk; 1176 total = 147 blocks x 8

// ------------------------------ router -------------------------------------
__global__ __launch_bounds__(256) void lomoe_router(
    const float* __restrict__ x, const float* __restrict__ W_r1,
    const float* __restrict__ b_r1, const float* __restrict__ W_r2,
    const float* __restrict__ b_r2, float* __restrict__ wcomb,
    float* __restrict__ probs_out) {
  __shared__ float pooled_s[Dn];
  __shared__ float h_s[Hn];
  __shared__ float logit_s[En];
  const int b = blockIdx.x;
  const int tid = threadIdx.x;
  const int wid = tid >> 5;
  const int lane = tid & 31;
  const float* xb = x + (size_t)b * Vn * Dn * Pn;

  // pooled[d] = mean over v,p ; one wave per d, lanes cover p (coalesced)
  for (int d = wid; d < Dn; d += 8) {
    float s = 0.f;
    const float* xd = xb + (size_t)d * Pn;
    for (int v = 0; v < Vn; ++v) {
      const float* xv = xd + (size_t)v * Dn * Pn;
      s += xv[lane] + xv[lane + 32];
    }
    for (int off = 16; off > 0; off >>= 1) s += __shfl_down(s, off, 32);
    if (lane == 0) pooled_s[d] = s * (1.0f / (float)(Vn * Pn));
  }
  __syncthreads();

  {  // h = gelu_exact(pooled @ W_r1^T + b_r1)
    const int j = tid;
    float s = b_r1[j];
    const float* w = W_r1 + (size_t)j * Dn;
    for (int d = 0; d < Dn; ++d) s += pooled_s[d] * w[d];
    h_s[j] = 0.5f * s * (1.0f + erff(s * 0.70710678118654752f));
  }
  __syncthreads();

  if (tid < En) {  // logits
    float s = b_r2[tid];
    const float* w = W_r2 + (size_t)tid * Hn;
    for (int j = 0; j < Hn; ++j) s += h_s[j] * w[j];
    logit_s[tid] = s;
  }
  __syncthreads();

  if (tid == 0) {  // softmax + top-2 + renormalize
    float mx = logit_s[0];
    for (int i = 1; i < En; ++i) mx = fmaxf(mx, logit_s[i]);
    float p[En], sum = 0.f;
    for (int i = 0; i < En; ++i) { p[i] = expf(logit_s[i] - mx); sum += p[i]; }
    const float inv = 1.0f / sum;
    for (int i = 0; i < En; ++i) { p[i] *= inv; probs_out[b * En + i] = p[i]; }
    int i0 = 0;
    for (int i = 1; i < En; ++i) if (p[i] > p[i0]) i0 = i;
    int i1 = (i0 == 0) ? 1 : 0;
    for (int i = 0; i < En; ++i) if (i != i0 && p[i] > p[i1]) i1 = i;
    const float denom = fmaxf(p[i0] + p[i1], 1e-6f);
    for (int i = 0; i < En; ++i) wcomb[b * En + i] = 0.f;
    wcomb[b * En + i0] = p[i0] / denom;
    wcomb[b * En + i1] = p[i1] / denom;
  }
}

// ------------------------------ fused GEMM ---------------------------------
// Cpart[chunk][672,848] = x_flat[672, Kchunk] x [W_base ; lora_A]^T (chunk k's)
// 32x64 wave tile: 2 A frags x 4 B frags = 8 f32 WMMAs per K-step of 4.
__global__ __launch_bounds__(256) void lomoe_gemm(
    const float* __restrict__ A, const float* __restrict__ Wb,
    const float* __restrict__ lA, float* __restrict__ Cpart) {
  const int tid = threadIdx.x;
  const int lane = tid & 31;
  int job = blockIdx.x * 8 + (tid >> 5);
  if (job >= JOBS * KCHUNKS) return;     // wave-uniform exit
  const int chunk = job / JOBS;
  job -= chunk * JOBS;
  const int ntile = job % NT2;
  const int mtile = job / NT2;
  const int m0 = mtile * 32;
  const int n0 = ntile * 64;
  const int k0 = chunk * KC;

  // frag layout: lanes 0-15 = rows 0..15 with K={0,1}; lanes 16-31 = same
  // rows with K={2,3}; one contiguous 8B load per lane per fragment.
  const int mloc = lane & 15;
  const int kb = (lane >> 4) << 1;       // 0 or 2

  const float* pA0 = A + (size_t)(m0 + mloc) * INn + k0 + kb;
  const float* pA1 = pA0 + (size_t)16 * INn;

  const float* pB[4];
#pragma unroll
  for (int j = 0; j < 4; ++j) {
    int n = n0 + 16 * j + mloc;
    if (n > NTOT - 1) n = NTOT - 1;      // tail frags: clamp, stores skipped
    pB[j] = (n < OUTn ? Wb + (size_t)n * INn
                      : lA + (size_t)(n - OUTn) * INn) + k0 + kb;
  }

  v8f acc[2][4] = {};

#pragma unroll 4
  for (int k = 0; k < KC; k += 4) {
    v2f a0 = *(const v2f*)(pA0 + k);
    v2f a1 = *(const v2f*)(pA1 + k);
    v2f bf[4];
#pragma unroll
    for (int j = 0; j < 4; ++j) bf[j] = *(const v2f*)(pB[j] + k);
#pragma unroll
    for (int j = 0; j < 4; ++j) {
      acc[0][j] = __builtin_amdgcn_wmma_f32_16x16x4_f32(
          false, a0, false, bf[j], (short)0, acc[0][j], false, false);
      acc[1][j] = __builtin_amdgcn_wmma_f32_16x16x4_f32(
          false, a1, false, bf[j], (short)0, acc[1][j], false, false);
    }
  }

  // C/D layout: VGPR r -> M = r (lanes 0-15) / r+8 (lanes 16-31), N = lane&15
  float* Cp = Cpart + (size_t)chunk * Mn * NTOT;
  const int nloc = lane & 15;
  const int mhalf = (lane >> 4) << 3;    // 0 or 8
#pragma unroll
  for (int j = 0; j < 4; ++j) {
    if (n0 + 16 * j < NTOT) {            // wave-uniform tail predicate
#pragma unroll
      for (int r = 0; r < 8; ++r) {
        Cp[(size_t)(m0 + mhalf + r) * NTOT + (n0 + 16 * j + nloc)] = acc[0][j][r];
        Cp[(size_t)(m0 + 16 + mhalf + r) * NTOT + (n0 + 16 * j + nloc)] = acc[1][j][r];
      }
    }
  }
}

// ------------------------------ epilogue -----------------------------------
// Deterministic split-K reduction + bias + routed LoRA combine.
__global__ __launch_bounds__(256) void lomoe_epilogue(
    const float* __restrict__ Cpart, const float* __restrict__ b_base,
    const float* __restrict__ lora_B, const float* __restrict__ wcomb,
    float* __restrict__ out) {
  __shared__ float low_s[NLORA];
  __shared__ float w_s[En];
  const int bv = blockIdx.x;
  const int b = bv / Vn;
  const int tid = threadIdx.x;
  if (tid < NLORA) {
    float s = 0.f;
#pragma unroll
    for (int c = 0; c < KCHUNKS; ++c)
      s += Cpart[(size_t)c * Mn * NTOT + (size_t)bv * NTOT + OUTn + tid];
    low_s[tid] = s;
  }
  if (tid < En) w_s[tid] = wcomb[b * En + tid] * SCALING;
  __syncthreads();
  for (int o = tid; o < OUTn; o += 256) {
    float acc = b_base[o];
#pragma unroll
    for (int c = 0; c < KCHUNKS; ++c)
      acc += Cpart[(size_t)c * Mn * NTOT + (size_t)bv * NTOT + o];
    for (int e = 0; e < En; ++e) {
      const float we = w_s[e];
      if (we != 0.0f) {                  // only top-k experts contribute
        const float* lb = lora_B + ((size_t)e * OUTn + o) * Rn;
        float d = 0.f;
#pragma unroll
        for (int r = 0; r < Rn; ++r) d += low_s[e * Rn + r] * lb[r];
        acc += we * d;
      }
    }
    out[(size_t)bv * OUTn + o] = acc;
  }
}

// ------------------------------ launch -------------------------------------
extern "C" void kernel_launch(void* const* d_in, const int* in_sizes, int n_in,
                              void* d_out, int out_size, void* d_ws,
                              size_t ws_size, hipStream_t stream) {
  const float* x      = (const float*)d_in[0];
  const float* W_base = (const float*)d_in[1];
  const float* b_base = (const float*)d_in[2];
  const float* W_r1   = (const float*)d_in[3];
  const float* b_r1   = (const float*)d_in[4];
  const float* W_r2   = (const float*)d_in[5];
  const float* b_r2   = (const float*)d_in[6];
  const float* lora_A = (const float*)d_in[7];
  const float* lora_B = (const float*)d_in[8];

  float* out   = (float*)d_out;                  // [B,V,OUT] then probs [B,E]
  float* probs = out + (size_t)Bn * Vn * OUTn;

  float* wcomb = (float*)d_ws;                   // [B,E]
  float* Cpart = wcomb + Bn * En;                // [KCHUNKS][672, 848] ~9.1 MB

  lomoe_router<<<Bn, 256, 0, stream>>>(x, W_r1, b_r1, W_r2, b_r2, wcomb, probs);
  lomoe_gemm<<<(JOBS * KCHUNKS + 7) / 8, 256, 0, stream>>>(x, W_base, lora_A, Cpart);
  lomoe_epilogue<<<Mn, 256, 0, stream>>>(Cpart, b_base, lora_B, wcomb, out);
}